// RGCN_6296422056695
// MI455X (gfx1250) — compile-verified
//
#include <hip/hip_runtime.h>

typedef float v2f __attribute__((ext_vector_type(2)));
typedef float v8f __attribute__((ext_vector_type(8)));

static constexpr int DIM    = 128;
static constexpr int LDSROW = 130;  // even pad: 8B-aligned v2f loads, conflict-free banks

__device__ __forceinline__ void atomAddF(float* p, float v) {
  unsafeAtomicAdd(p, v);  // hardware global_atomic_add_f32 (no CAS loop)
}

// ---------------------------------------------------------------- utilities
__global__ void zero_f32(float* __restrict__ p, int n) {
  int i = blockIdx.x * blockDim.x + threadIdx.x;
  if (i < n) p[i] = 0.0f;
}

__global__ void deg_accum(const int* __restrict__ src, const int* __restrict__ dst,
                          float* __restrict__ dout, float* __restrict__ din, int e) {
  int i = blockIdx.x * blockDim.x + threadIdx.x;
  if (i < e) {
    atomAddF(&dout[src[i]], 1.0f);
    atomAddF(&din[dst[i]], 1.0f);
  }
}

__global__ void deg_finalize(float* __restrict__ p, int n) {
  int i = blockIdx.x * blockDim.x + threadIdx.x;
  if (i < n) {
    float d = p[i];
    d = d < 1.0f ? 1.0f : d;          // jnp.clip(deg, 1.0)
    p[i] = rsqrtf(d);                 // deg^-1/2
  }
}

// out[i] = b0[i%128] (+ b1[i%128]); pre-seeds the relation-sum accumulator
__global__ void bias_init2(float* __restrict__ out, const float* __restrict__ b0,
                           const float* __restrict__ b1, int n) {
  int i = blockIdx.x * blockDim.x + threadIdx.x;
  if (i < n) {
    float v = b0[i & (DIM - 1)];
    if (b1) v += b1[i & (DIM - 1)];
    out[i] = v;
  }
}

// ---------------------------------------------------------------- dense GEMM
// H[r,:] = (relu?(X[r,:])) * rowScale[r] @ Wm   for a 16-row tile per block.
// 8 waves/block, each wave owns one 16-column slice; K=128 via 32x WMMA f32 16x16x4.
__global__ void __launch_bounds__(256)
gemm_rowscale_wmma(const float* __restrict__ X, const float* __restrict__ Wm,
                   const float* __restrict__ rowScale, float* __restrict__ H,
                   int nRows, int relu) {
  __shared__ float As[16 * LDSROW];
  const int tid  = threadIdx.x;
  const int wave = tid >> 5;
  const int lane = tid & 31;
  const int row0 = blockIdx.x << 4;

  // Stage 16x128 A tile in LDS with fused relu + deg_out^-1/2 row scaling.
#pragma unroll
  for (int it = 0; it < 8; ++it) {
    int idx = tid + it * 256;              // 0..2047, coalesced
    int r = idx >> 7, c = idx & (DIM - 1);
    int gr = row0 + r;
    float v = 0.0f;
    if (gr < nRows) {
      v = X[(size_t)gr * DIM + c];
      if (relu) v = v > 0.0f ? v : 0.0f;
      v *= rowScale[gr];
    }
    As[r * LDSROW + c] = v;
  }
  __syncthreads();

  const int m    = lane & 15;              // A row / B column / D column
  const int khi  = (lane >> 4) << 1;       // lanes 16-31 hold K+2,K+3
  const int col0 = wave << 4;
  const v2f* arow = reinterpret_cast<const v2f*>(&As[m * LDSROW]);

  v8f acc = {};
#pragma unroll
  for (int k0 = 0; k0 < DIM; k0 += 4) {
    v2f a = arow[(k0 + khi) >> 1];         // A[m][k], A[m][k+1] (ds_load_b64)
    v2f b;
    b.x = Wm[(k0 + khi) * DIM + col0 + m];
    b.y = Wm[(k0 + khi + 1) * DIM + col0 + m];
    acc = __builtin_amdgcn_wmma_f32_16x16x4_f32(false, a, false, b, (short)0, acc,
                                                false, false);
  }

  // D layout: lanes 0-15 -> rows v, lanes 16-31 -> rows v+8; col = col0+m
  const int rbase = row0 + ((lane >> 4) << 3);
#pragma unroll
  for (int v = 0; v < 8; ++v) {
    int gr = rbase + v;
    if (gr < nRows) H[(size_t)gr * DIM + col0 + m] = acc[v];
  }
}

// ---------------------------------------------------------------- edge scatter
// out[dst] += h[src] * deg_in_scale[dst]; one wave per edge, float4 per lane.
__global__ void __launch_bounds__(256)
scatter_edges(const float* __restrict__ h, const int* __restrict__ src,
              const int* __restrict__ dst, const float* __restrict__ dinScale,
              float* __restrict__ out, int e) {
  int wave = blockIdx.x * 8 + (threadIdx.x >> 5);
  int lane = threadIdx.x & 31;
  if (wave >= e) return;
  int s = src[wave];
  int d = dst[wave];
  float sc = dinScale[d];
  const float4 v = *reinterpret_cast<const float4*>(h + (size_t)s * DIM + lane * 4);
  float* o = out + (size_t)d * DIM + lane * 4;
  atomAddF(o + 0, v.x * sc);
  atomAddF(o + 1, v.y * sc);
  atomAddF(o + 2, v.z * sc);
  atomAddF(o + 3, v.w * sc);
}

// ---------------------------------------------------------------- driver
extern "C" void kernel_launch(void* const* d_in, const int* in_sizes, int n_in,
                              void* d_out, int out_size, void* d_ws, size_t ws_size,
                              hipStream_t stream) {
  // Input order: x_A, x_B, then {W,b} for l=1..3 x r=r1..r3, then src/dst per relation.
  const float* xA = (const float*)d_in[0];
  const float* xB = (const float*)d_in[1];
  const float* Wm[3][3];
  const float* Bv[3][3];
  for (int l = 0; l < 3; ++l)
    for (int r = 0; r < 3; ++r) {
      Wm[l][r] = (const float*)d_in[2 + l * 6 + r * 2];
      Bv[l][r] = (const float*)d_in[3 + l * 6 + r * 2];
    }
  const int* srcs[3];
  const int* dsts[3];
  int E[3];
  for (int r = 0; r < 3; ++r) {
    srcs[r] = (const int*)d_in[20 + 2 * r];
    dsts[r] = (const int*)d_in[21 + 2 * r];
    E[r]    = in_sizes[20 + 2 * r];
  }
  const int Nn = in_sizes[0] / DIM;  // 100000
  const size_t ND = (size_t)Nn * DIM;

  // Workspace: 6 degree-scale arrays, one h buffer, ping-pong layer buffers.
  float* ws = (float*)d_ws;
  float* so[3];
  float* si[3];
  for (int r = 0; r < 3; ++r) {
    so[r] = ws + (size_t)(2 * r) * Nn;      // deg_out scale (source side)
    si[r] = ws + (size_t)(2 * r + 1) * Nn;  // deg_in  scale (dest side)
  }
  float* h  = ws + (size_t)6 * Nn;
  float* A0 = h + ND;
  float* B0 = A0 + ND;
  float* A1 = B0 + ND;
  float* B1 = A1 + ND;

  float* outA = (float*)d_out;       // final hA -> d_out[0 .. N*D)
  float* outB = outA + ND;           // final hB -> d_out[N*D .. 2N*D)

  const int T = 256;

  // Degree scales (shared by all layers).
  zero_f32<<<(6 * Nn + T - 1) / T, T, 0, stream>>>(ws, 6 * Nn);
  for (int r = 0; r < 3; ++r)
    deg_accum<<<(E[r] + T - 1) / T, T, 0, stream>>>(srcs[r], dsts[r], so[r], si[r], E[r]);
  deg_finalize<<<(6 * Nn + T - 1) / T, T, 0, stream>>>(ws, 6 * Nn);

  const float* inA[3] = { xA, A0, A1 };
  const float* inB[3] = { xB, B0, B1 };
  float* oA[3] = { A0, A1, outA };
  float* oB[3] = { B0, B1, outB };

  const int mtiles  = (Nn + 15) / 16;
  const int ndElems = (int)ND;

  for (int l = 0; l < 3; ++l) {
    const int relu = (l > 0) ? 1 : 0;  // relu fused into GEMM input for layers 2,3
    // outA accumulates r2 (B->A) + r3 (A->A); outB accumulates r1 (A->B).
    bias_init2<<<(ndElems + T - 1) / T, T, 0, stream>>>(oA[l], Bv[l][1], Bv[l][2], ndElems);
    bias_init2<<<(ndElems + T - 1) / T, T, 0, stream>>>(oB[l], Bv[l][0], nullptr, ndElems);

    // r1: A -> B
    gemm_rowscale_wmma<<<mtiles, 256, 0, stream>>>(inA[l], Wm[l][0], so[0], h, Nn, relu);
    scatter_edges<<<(E[0] + 7) / 8, 256, 0, stream>>>(h, srcs[0], dsts[0], si[0], oB[l], E[0]);
    // r2: B -> A
    gemm_rowscale_wmma<<<mtiles, 256, 0, stream>>>(inB[l], Wm[l][1], so[1], h, Nn, relu);
    scatter_edges<<<(E[1] + 7) / 8, 256, 0, stream>>>(h, srcs[1], dsts[1], si[1], oA[l], E[1]);
    // r3: A -> A
    gemm_rowscale_wmma<<<mtiles, 256, 0, stream>>>(inA[l], Wm[l][2], so[2], h, Nn, relu);
    scatter_edges<<<(E[2] + 7) / 8, 256, 0, stream>>>(h, srcs[2], dsts[2], si[2], oA[l], E[2]);
  }

  (void)n_in; (void)out_size; (void)ws_size;
}